// RobustContrastNormalization_65652870086685
// MI455X (gfx1250) — compile-verified
//
#include <hip/hip_runtime.h>
#include <stdint.h>
#include <stddef.h>

// Problem geometry (fixed by the reference setup).
#define B_SAMPLES 256
#define HPIX 384
#define WPIX 384
#define NCH 3
#define NPIX (HPIX * WPIX)                 // 147456 pixels per sample
#define TOTAL_PIX (B_SAMPLES * NPIX)       // 37,748,736
#define PIX4_PER_SAMPLE (NPIX / 4)         // 36864
#define MEAN_BLOCK 256
#define BLOCKS_PER_SAMPLE (PIX4_PER_SAMPLE / MEAN_BLOCK)  // 144

// Radix-select config: 8-bit fused pass 0, then two 12-bit streamed passes.
#define SEL_THREADS 256
#define CHUNK_FLOATS 2048                  // 8 KB per staging buffer
#define SEGS_PER_THREAD (CHUNK_FLOATS / (SEL_THREADS * 4))  // 2 x b128 per thread
#define NCHUNKS (NPIX / CHUNK_FLOATS)      // 72
#define NBUF 4                             // DMA ring depth (3 chunks in flight)
#define NRANK 4
#define HBINS 4096                         // 12-bit histogram
#define GROUP (HBINS / SEL_THREADS)        // 16 bins per group-sum lane

static_assert(NPIX % CHUNK_FLOATS == 0, "chunking must tile the sample");
static_assert(SEGS_PER_THREAD == 2, "wait-count immediates assume 2 segs");
static_assert(NCHUNKS >= NBUF, "prologue assumes at least NBUF chunks");

typedef float f32x4 __attribute__((ext_vector_type(4)));

// ---- CDNA5 async global->LDS DMA (ASYNCcnt path) ----------------------------
#define WAIT_ASYNC(n) asm volatile("s_wait_asynccnt " #n ::: "memory")
#define WAIT_DS0()    asm volatile("s_wait_dscnt 0" ::: "memory")

__device__ __forceinline__ void async_copy16(unsigned lds_off, const void* gptr) {
  unsigned long long ga = (unsigned long long)(uintptr_t)gptr;
  // Per-lane 16B copy from global to LDS, tracked by ASYNCcnt.
  asm volatile("global_load_async_to_lds_b128 %0, %1, off"
               :: "v"(lds_off), "v"(ga)
               : "memory");
}

__device__ __forceinline__ void issue_chunk(const float* gbase, unsigned lds_base, int tid) {
#pragma unroll
  for (int s = 0; s < SEGS_PER_THREAD; ++s) {
    const int seg = s * SEL_THREADS + tid;
    async_copy16(lds_base + (unsigned)(seg * 16), (const char*)gbase + (size_t)seg * 16);
  }
}

// ---- monotonic float <-> u32 key mapping ------------------------------------
__device__ __forceinline__ unsigned f2mono(float f) {
  unsigned u = __float_as_uint(f);
  return u ^ ((u >> 31) ? 0xFFFFFFFFu : 0x80000000u);
}
__device__ __forceinline__ float mono2f(unsigned m) {
  return __uint_as_float(m ^ ((m >> 31) ? 0x80000000u : 0xFFFFFFFFu));
}

// ---- kernel 0: zero the per-sample pass-0 histograms ------------------------
__global__ void zero_kernel(unsigned* __restrict__ p, int n) {
  int i = blockIdx.x * blockDim.x + threadIdx.x;
  if (i < n) p[i] = 0u;
}

// ---- kernel 1: channel mean + fused radix pass-0 (top byte) histogram -------
// x is 453 MB touched exactly once -> non-temporal loads so it cannot evict
// the 151 MB xbar working set from the 192 MB L2. xbar stores stay RT so the
// select/normalize passes are L2 hits.
__global__ __launch_bounds__(MEAN_BLOCK)
void mean_hist_kernel(const float* __restrict__ x, float* __restrict__ xg,
                      unsigned* __restrict__ ghist) {
  __shared__ unsigned h[256];
  const int tid = threadIdx.x;
  const int blk = blockIdx.x;
  const int b   = blk / BLOCKS_PER_SAMPLE;

  h[tid] = 0u;
  __syncthreads();

  const int i4 = blk * MEAN_BLOCK + tid;   // float4 output index (12 input floats)
  const f32x4* x4 = (const f32x4*)x;
  f32x4 a0 = __builtin_nontemporal_load(x4 + 3 * (size_t)i4 + 0);
  f32x4 a1 = __builtin_nontemporal_load(x4 + 3 * (size_t)i4 + 1);
  f32x4 a2 = __builtin_nontemporal_load(x4 + 3 * (size_t)i4 + 2);
  const float inv3 = 1.0f / 3.0f;
  f32x4 m;
  m.x = (a0.x + a0.y + a0.z) * inv3;
  m.y = (a0.w + a1.x + a1.y) * inv3;
  m.z = (a1.z + a1.w + a2.x) * inv3;
  m.w = (a2.y + a2.z + a2.w) * inv3;
  ((f32x4*)xg)[i4] = m;

  atomicAdd(&h[f2mono(m.x) >> 24], 1u);
  atomicAdd(&h[f2mono(m.y) >> 24], 1u);
  atomicAdd(&h[f2mono(m.z) >> 24], 1u);
  atomicAdd(&h[f2mono(m.w) >> 24], 1u);
  __syncthreads();

  unsigned c = h[tid];
  if (c) atomicAdd(&ghist[(size_t)b * 256 + tid], c);
}

// ---- kernel 2: 4-rank radix select (8 + 12 + 12 bits) -----------------------
// One block per sample; selects sorted[k_lo], sorted[k_lo+1], sorted[k_hi],
// sorted[k_hi+1] exactly (duplicate-safe) with two async-streamed passes.
__global__ __launch_bounds__(SEL_THREADS)
void select_kernel(const float* __restrict__ xg, const unsigned* __restrict__ ghist,
                   float* __restrict__ stats,
                   int k_lo, float fr_lo, int k_hi, float fr_hi) {
  __shared__ f32x4    stage[NBUF][CHUNK_FLOATS / 4]; // 32 KB ring (3 in flight)
  __shared__ unsigned hist[NRANK][HBINS];            // 64 KB (legal: 320 KB/WGP)
  __shared__ unsigned gsum[NRANK][SEL_THREADS];      // 4 KB group sums
  __shared__ unsigned sh_prefix[NRANK];
  __shared__ unsigned sh_krem[NRANK];

  const int tid = threadIdx.x;
  const int b   = blockIdx.x;
  const float* src = xg + (size_t)b * NPIX;

  // Pass 0 (top 8 bits) resolved from the precomputed per-sample histogram.
  gsum[0][tid] = ghist[(size_t)b * 256 + tid];
  __syncthreads();
  if (tid < NRANK) {
    const unsigned target = (tid == 0) ? (unsigned)k_lo
                          : (tid == 1) ? (unsigned)(k_lo + 1)
                          : (tid == 2) ? (unsigned)k_hi
                                       : (unsigned)(k_hi + 1);
    unsigned cum = 0, bsel = 255, krem = target;
    for (int bin = 0; bin < 256; ++bin) {
      unsigned c = gsum[0][bin];
      if (cum + c > target) { bsel = (unsigned)bin; krem = target - cum; break; }
      cum += c;
    }
    sh_prefix[tid] = bsel << 24;
    sh_krem[tid]   = krem;
  }
  __syncthreads();

  unsigned offs[NBUF];
#pragma unroll
  for (int i = 0; i < NBUF; ++i) offs[i] = (unsigned)(uintptr_t)&stage[i][0];
  unsigned* hf = &hist[0][0];

  for (int pass = 0; pass < 2; ++pass) {
    const int shift = pass ? 0 : 12;
    const unsigned himask = pass ? 0xFFFFF000u : 0xFF000000u;
    const unsigned pfx0 = sh_prefix[0], pfx1 = sh_prefix[1];
    const unsigned pfx2 = sh_prefix[2], pfx3 = sh_prefix[3];
#pragma unroll
    for (int i = 0; i < (NRANK * HBINS) / SEL_THREADS; ++i)
      hf[i * SEL_THREADS + tid] = 0u;
    __syncthreads();

    // 4-deep ring, 3 chunks (6 async ops) in flight per wave. Each lane only
    // consumes segments it issued -> no per-chunk barrier, just counter waits.
    issue_chunk(src + 0 * CHUNK_FLOATS, offs[0], tid);
    issue_chunk(src + 1 * CHUNK_FLOATS, offs[1], tid);
    issue_chunk(src + 2 * CHUNK_FLOATS, offs[2], tid);
    for (int c = 0; c < NCHUNKS; ++c) {
      const int buf = c & (NBUF - 1);
      if (c + 3 < NCHUNKS) {
        WAIT_DS0();        // recycled buffer's LDS reads fully retired
        issue_chunk(src + (size_t)(c + 3) * CHUNK_FLOATS, offs[(c + 3) & (NBUF - 1)], tid);
        WAIT_ASYNC(6);     // chunks c+1..c+3 may remain in flight; chunk c done
      } else if (c + 3 == NCHUNKS) {
        WAIT_ASYNC(4);
      } else if (c + 2 == NCHUNKS) {
        WAIT_ASYNC(2);
      } else {
        WAIT_ASYNC(0);
      }
#pragma unroll
      for (int s = 0; s < SEGS_PER_THREAD; ++s) {
        const f32x4 v = stage[buf][s * SEL_THREADS + tid];
        unsigned kk[4] = { f2mono(v.x), f2mono(v.y), f2mono(v.z), f2mono(v.w) };
#pragma unroll
        for (int j = 0; j < 4; ++j) {
          const unsigned k   = kk[j];
          const unsigned bin = (k >> shift) & (HBINS - 1);
          if ((k & himask) == pfx0) atomicAdd(&hist[0][bin], 1u);
          if ((k & himask) == pfx1) atomicAdd(&hist[1][bin], 1u);
          if ((k & himask) == pfx2) atomicAdd(&hist[2][bin], 1u);
          if ((k & himask) == pfx3) atomicAdd(&hist[3][bin], 1u);
        }
      }
    }
    WAIT_DS0();
    __syncthreads();

    // Two-level bucket search: parallel 16-bin group sums, then short serial
    // scans (<= 256 + 16 steps) per rank instead of a 4096-step serial scan.
    for (int r = 0; r < NRANK; ++r) {
      unsigned s = 0;
#pragma unroll
      for (int j = 0; j < GROUP; ++j) s += hist[r][tid * GROUP + j];
      gsum[r][tid] = s;
    }
    __syncthreads();
    if (tid < NRANK) {
      const unsigned krem = sh_krem[tid];
      unsigned cum = 0;
      int g = SEL_THREADS - 1;
      for (int i = 0; i < SEL_THREADS; ++i) {
        unsigned c = gsum[tid][i];
        if (cum + c > krem) { g = i; break; }
        cum += c;
      }
      unsigned bsel = (unsigned)(g * GROUP + GROUP - 1);
      for (int i = g * GROUP; i < (g + 1) * GROUP; ++i) {
        unsigned c = hist[tid][i];
        if (cum + c > krem) { bsel = (unsigned)i; break; }
        cum += c;
      }
      sh_prefix[tid] |= bsel << shift;
      sh_krem[tid]    = krem - cum;
    }
    __syncthreads();
  }

  if (tid == 0) {
    const float s_lo0 = mono2f(sh_prefix[0]);
    const float s_lo1 = mono2f(sh_prefix[1]);
    const float s_hi0 = mono2f(sh_prefix[2]);
    const float s_hi1 = mono2f(sh_prefix[3]);
    const float lo  = s_lo0 + fr_lo * (s_lo1 - s_lo0);   // linear-interp quantile
    const float hi  = s_hi0 + fr_hi * (s_hi1 - s_hi0);
    const float rng = fmaxf(hi - lo, 1e-6f);
    stats[2 * b + 0] = lo;
    stats[2 * b + 1] = rng;
  }
}

// ---- kernel 3: in-place normalize (d_out holds xbar, L2-resident) -----------
__global__ __launch_bounds__(256)
void normalize_kernel(float* __restrict__ xg_out, const float* __restrict__ stats) {
  const int i = blockIdx.x * blockDim.x + threadIdx.x;    // float4 index, exact grid
  const int b = i / PIX4_PER_SAMPLE;
  const float lo  = stats[2 * b + 0];
  const float rng = stats[2 * b + 1];
  f32x4 v = ((const f32x4*)xg_out)[i];
  f32x4 o;
  o.x = fminf(fmaxf((v.x - lo) / rng, 0.0f), 1.0f);
  o.y = fminf(fmaxf((v.y - lo) / rng, 0.0f), 1.0f);
  o.z = fminf(fmaxf((v.z - lo) / rng, 0.0f), 1.0f);
  o.w = fminf(fmaxf((v.w - lo) / rng, 0.0f), 1.0f);
  ((f32x4*)xg_out)[i] = o;
}

extern "C" void kernel_launch(void* const* d_in, const int* in_sizes, int n_in,
                              void* d_out, int out_size, void* d_ws, size_t ws_size,
                              hipStream_t stream) {
  (void)in_sizes; (void)n_in; (void)out_size; (void)ws_size;
  const float* x = (const float*)d_in[0];
  float* out = (float*)d_out;                       // reused as the xbar buffer
  unsigned* ghist = (unsigned*)d_ws;                // 256 samples * 256 bins
  float* stats = (float*)(ghist + (size_t)B_SAMPLES * 256);  // {lo, rng} per sample

  const int nh = B_SAMPLES * 256;
  zero_kernel<<<(nh + 255) / 256, 256, 0, stream>>>(ghist, nh);

  mean_hist_kernel<<<B_SAMPLES * BLOCKS_PER_SAMPLE, MEAN_BLOCK, 0, stream>>>(x, out, ghist);

  const double pl = (double)(NPIX - 1) * 0.10;      // 14745.5 -> k=14745, fr=0.5
  const double ph = (double)(NPIX - 1) * 0.90;      // 132709.5 -> k=132709, fr=0.5
  const int   k_lo = (int)pl,           k_hi = (int)ph;
  const float fr_lo = (float)(pl - k_lo), fr_hi = (float)(ph - k_hi);

  select_kernel<<<B_SAMPLES, SEL_THREADS, 0, stream>>>(out, ghist, stats,
                                                       k_lo, fr_lo, k_hi, fr_hi);

  normalize_kernel<<<TOTAL_PIX / 4 / 256, 256, 0, stream>>>(out, stats);
}